// GCN_61907658605175
// MI455X (gfx1250) — compile-verified
//
#include <hip/hip_runtime.h>
#include <hip/hip_bf16.h>

#define IN_FEATS 16
#define KF 48   // K * IN_FEATS
#define OUT_FEATS 2

typedef __attribute__((ext_vector_type(2))) float v2f;
typedef __attribute__((ext_vector_type(8))) float v8f;

// Native global_atomic_add_f32 (L2-resident targets -> fast path), no CAS loop.
__device__ __forceinline__ void atomic_add_f32(float* p, float v) {
    __hip_atomic_fetch_add(p, v, __ATOMIC_RELAXED, __HIP_MEMORY_SCOPE_AGENT);
}

// re = 2/lambda via single v_rcp_f32 (no IEEE div expansion needed here).
__device__ __forceinline__ float re_scale(const float* __restrict__ lambda_max) {
    return 2.0f * __builtin_amdgcn_rcpf(lambda_max[0]);
}

// ---------------- zero workspace (norm + S accumulator) ----------------
__global__ void gcn_zero_f4(float4* __restrict__ p, int n4) {
    int i = blockIdx.x * blockDim.x + threadIdx.x;
    if (i < n4) p[i] = make_float4(0.f, 0.f, 0.f, 0.f);
}

// ---------------- degree: deg[dst] += 1 ----------------
__global__ void gcn_degree(const long long* __restrict__ dst,
                           float* __restrict__ deg, int E) {
    int e = blockIdx.x * blockDim.x + threadIdx.x;
    if (e < E) atomic_add_f32(&deg[(int)dst[e]], 1.0f);
}

// ---------------- norm = rsqrt(max(deg,1)) in place ----------------
__global__ void gcn_norm(float* __restrict__ deg, int N) {
    int i = blockIdx.x * blockDim.x + threadIdx.x;
    if (i < N) deg[i] = rsqrtf(fmaxf(deg[i], 1.0f));
}

// ---------------- prop: S[dst] += X[src]*norm[src] ----------------
// One thread per edge: indices loaded once (coalesced), 4x float4 gather,
// 16 f32 atomics into the L2-resident accumulator.
__global__ void gcn_prop(const float* __restrict__ X,
                         const float* __restrict__ norm,
                         const long long* __restrict__ src,
                         const long long* __restrict__ dst,
                         float* __restrict__ S, int E) {
    int e = blockIdx.x * blockDim.x + threadIdx.x;
    if (e >= E) return;
    int s = (int)src[e];
    int d = (int)dst[e];
    float ns = norm[s];
    const float4* xr = reinterpret_cast<const float4*>(&X[s * IN_FEATS]);
    float* base = &S[d * IN_FEATS];
    #pragma unroll
    for (int q = 0; q < 4; ++q) {
        float4 x = xr[q];
        atomic_add_f32(base + q * 4 + 0, x.x * ns);
        atomic_add_f32(base + q * 4 + 1, x.y * ns);
        atomic_add_f32(base + q * 4 + 2, x.z * ns);
        atomic_add_f32(base + q * 4 + 3, x.w * ns);
    }
}

// ---------------- X1 = -re*(S*norm_dst) + X0*(re-1);  S <- 0 (fused) -------
__global__ void gcn_x1(const float* __restrict__ X0,
                       float* __restrict__ S,
                       const float* __restrict__ norm,
                       const float* __restrict__ lambda_max,
                       float* __restrict__ X1, int total) {
    int i = blockIdx.x * blockDim.x + threadIdx.x;
    if (i >= total) return;
    float re = re_scale(lambda_max);
    float p = S[i] * norm[i >> 4];   // IN_FEATS == 16
    X1[i] = -re * p + X0[i] * (re - 1.0f);
    S[i] = 0.0f;                     // re-zero for the second prop
}

// ---------------- projection: relu([X0|X1|X2] @ W^T) via V_WMMA_F32_16X16X4_F32
// One wave per 16-node tile. K=48 -> 12 chunks of 4. X2 computed on the fly:
//   X2 = -2*re*(S*norm) + 2*(re-1)*X1 - X0
// W is staged zero-padded (16x48) in LDS so B loads are branch-free ds_load_b64.
__global__ void gcn_project_wmma(const float* __restrict__ X0,
                                 const float* __restrict__ X1,
                                 const float* __restrict__ S,
                                 const float* __restrict__ norm,
                                 const float* __restrict__ W,          // (2,48)
                                 const float* __restrict__ lambda_max,
                                 float* __restrict__ out,              // (N,2)
                                 int N, int n_tiles) {
    __shared__ float Wlds[16 * KF];  // zero-padded B: rows 2..15 are zero

    // Uniform fill + barrier BEFORE any wave can exit.
    for (int i = threadIdx.x; i < 16 * KF; i += blockDim.x) {
        int row = i / KF;
        Wlds[i] = (row < OUT_FEATS) ? W[i] : 0.0f;
    }
    __syncthreads();

    int wave = (int)((blockIdx.x * blockDim.x + threadIdx.x) >> 5);
    int lane = threadIdx.x & 31;
    if (wave >= n_tiles) return;                 // wave-uniform: EXEC all-ones inside

    const float re = re_scale(lambda_max);
    const float c1 = re - 1.0f;

    // A layout (16x4 f32): lane = M (lanes 0-15 and 16-31 both hold M=0..15);
    // VGPR0 = K{0|2}, VGPR1 = K{1|3} depending on lane half.
    int mrow  = lane & 15;
    int khalf = lane >> 4;                       // 0 -> K=k0,k0+1 ; 1 -> K=k0+2,k0+3
    int node  = wave * 16 + mrow;
    int nodec = node < N ? node : (N - 1);       // clamp for tail tile (loads only)
    float nrm = norm[nodec];

    v8f acc = {};
    #pragma unroll
    for (int kc = 0; kc < 12; ++kc) {
        int k0  = kc * 4;
        int f   = (k0 & 15) + 2 * khalf;         // feat index within 16-feat block
        int idx = nodec * IN_FEATS + f;
        v2f a;
        if (kc < 4) {                            // X0 block
            a.x = X0[idx];     a.y = X0[idx + 1];
        } else if (kc < 8) {                     // X1 block
            a.x = X1[idx];     a.y = X1[idx + 1];
        } else {                                 // X2 block, formed inline
            a.x = -2.0f * re * S[idx]     * nrm + 2.0f * c1 * X1[idx]     - X0[idx];
            a.y = -2.0f * re * S[idx + 1] * nrm + 2.0f * c1 * X1[idx + 1] - X0[idx + 1];
        }
        // B layout (4x16 f32): lane = N column; VGPR0 = K{0|2}, VGPR1 = K{1|3}.
        // Unconditional, 8B-aligned LDS load (zero-padded rows) -> no EXEC games.
        v2f b = *reinterpret_cast<const v2f*>(&Wlds[mrow * KF + k0 + 2 * khalf]);
        acc = __builtin_amdgcn_wmma_f32_16x16x4_f32(
                  false, a, false, b, (short)0, acc, false, false);
    }

    // D layout (16x16 f32): lane = Ncol + 16*(M>=8); vgpr v holds M = 8*khalf + v.
    int ncol = lane & 15;
    if (ncol < OUT_FEATS) {
        int mbase = khalf * 8;
        #pragma unroll
        for (int v = 0; v < 8; ++v) {
            int n_out = wave * 16 + mbase + v;
            if (n_out < N)
                out[n_out * OUT_FEATS + ncol] = fmaxf(acc[v], 0.0f);
        }
    }
}

extern "C" void kernel_launch(void* const* d_in, const int* in_sizes, int n_in,
                              void* d_out, int out_size, void* d_ws, size_t ws_size,
                              hipStream_t stream) {
    (void)n_in; (void)out_size; (void)ws_size;

    const float*     in_feat = (const float*)d_in[0];       // (N,16) f32
    const float*     W       = (const float*)d_in[1];       // (2,48) f32
    const long long* src     = (const long long*)d_in[2];   // (E,)   i64
    const long long* dst     = (const long long*)d_in[3];   // (E,)   i64
    const float*     lambda  = (const float*)d_in[4];       // (1,)   f32
    float*           out     = (float*)d_out;               // (N,2)  f32

    const int N = in_sizes[0] / IN_FEATS;                   // 100000
    const int E = in_sizes[2];                              // 3200000
    const int F = N * IN_FEATS;

    // workspace layout (floats): [ norm:N | S:16N | X1:16N ]
    float* ws   = (float*)d_ws;
    float* norm = ws;
    float* S    = ws + N;
    float* X1   = ws + (size_t)N + (size_t)F;

    const int TB = 256;

    // 0) zero norm + S  (17N floats, contiguous, float4 stores)
    {
        int n4 = (N + F) / 4;
        gcn_zero_f4<<<(n4 + TB - 1) / TB, TB, 0, stream>>>((float4*)ws, n4);
    }
    // 1) degree
    gcn_degree<<<(E + TB - 1) / TB, TB, 0, stream>>>(dst, norm, E);
    // 2) norm = rsqrt(max(deg,1))
    gcn_norm<<<(N + TB - 1) / TB, TB, 0, stream>>>(norm, N);
    // 3) prop 1: S += (X0*norm)[src] scattered to dst
    gcn_prop<<<(E + TB - 1) / TB, TB, 0, stream>>>(in_feat, norm, src, dst, S, E);
    // 4) X1 = -re*S*norm + (re-1)*X0 ; S <- 0
    gcn_x1<<<(F + TB - 1) / TB, TB, 0, stream>>>(in_feat, S, norm, lambda, X1, F);
    // 5) prop 2: S += (X1*norm)[src] scattered to dst
    gcn_prop<<<(E + TB - 1) / TB, TB, 0, stream>>>(X1, norm, src, dst, S, E);
    // 6) WMMA projection + ReLU (X2 formed inline)
    {
        int n_tiles = (N + 15) / 16;
        int waves_per_block = TB / 32;
        int blocks = (n_tiles + waves_per_block - 1) / waves_per_block;
        gcn_project_wmma<<<blocks, TB, 0, stream>>>(in_feat, X1, S, norm, W, lambda,
                                                    out, N, n_tiles);
    }
}